// WCAContrastiveLoss_12189117186265
// MI455X (gfx1250) — compile-verified
//
#include <hip/hip_runtime.h>

#define TEMP_T 0.07f
#define INV_T  (1.0f / 0.07f)
#define EPS_F  1e-8f

typedef float v2f __attribute__((ext_vector_type(2)));
typedef float v8f __attribute__((ext_vector_type(8)));

constexpr int Bn = 256, Vn = 16, Dn = 1024;
constexpr int BV = Bn * Vn;                 // 4096 rows
constexpr int NTILES = BV / 16;             // 256 tiles of 16 rows
constexpr int LDS_STRIDE = Dn + 4;          // pad 4 floats -> distinct banks for row-strided reads

// ---------------------------------------------------------------------------
// Kernel 1: L2-normalize degraded rows (BV x D) and clean rows (B x D)
// ---------------------------------------------------------------------------
__global__ void k_normalize(const float* __restrict__ clean,
                            const float* __restrict__ degr,
                            float* __restrict__ nrm,
                            float* __restrict__ cn) {
    __shared__ float red[256];
    const int row = blockIdx.x;
    const float* src;
    float* dst;
    if (row < BV) {
        src = degr + (size_t)row * Dn;
        dst = nrm  + (size_t)row * Dn;
    } else {
        const int r = row - BV;
        src = clean + (size_t)r * Dn;
        dst = cn    + (size_t)r * Dn;
    }
    float s = 0.0f;
    for (int d = threadIdx.x; d < Dn; d += 256) {
        float x = src[d];
        s += x * x;
    }
    red[threadIdx.x] = s;
    __syncthreads();
    for (int off = 128; off > 0; off >>= 1) {
        if (threadIdx.x < off) red[threadIdx.x] += red[threadIdx.x + off];
        __syncthreads();
    }
    const float inv = 1.0f / sqrtf(red[0] + 1e-12f);
    for (int d = threadIdx.x; d < Dn; d += 256) dst[d] = src[d] * inv;
}

// ---------------------------------------------------------------------------
// Kernel 2: pos[b,v] = exp(dot(n[b,v], c[b]) / T)
// ---------------------------------------------------------------------------
__global__ void k_pos(const float* __restrict__ nrm,
                      const float* __restrict__ cn,
                      float* __restrict__ gpos) {
    __shared__ float red[256];
    const int b = blockIdx.x;
    const float* c = cn + (size_t)b * Dn;
    for (int v = 0; v < Vn; ++v) {
        const float* n = nrm + ((size_t)b * Vn + v) * Dn;
        float s = 0.0f;
        for (int d = threadIdx.x; d < Dn; d += 256) s += n[d] * c[d];
        red[threadIdx.x] = s;
        __syncthreads();
        for (int off = 128; off > 0; off >>= 1) {
            if (threadIdx.x < off) red[threadIdx.x] += red[threadIdx.x + off];
            __syncthreads();
        }
        if (threadIdx.x == 0) gpos[b * Vn + v] = expf(red[0] * INV_T);
        __syncthreads();
    }
}

// ---------------------------------------------------------------------------
// Kernel 3 (WMMA): gram-matrix row sums + jaccard-weighted sibling sums.
// One workgroup (8 waves) handles TWO i-tiles (32 rows) staged in LDS
// (~128KB of the WGP's 320KB). Each wave sweeps 32 of the 256 j-tiles (stride 8); every
// B fragment loaded from global is reused by two v_wmma_f32_16x16x4_f32
// (one per i-tile), halving L2 traffic per FLOP.
// ---------------------------------------------------------------------------
__global__ void __launch_bounds__(256) k_gram(const float* __restrict__ nrm,
                                              const float* __restrict__ w,
                                              const unsigned char* __restrict__ mask,
                                              float* __restrict__ grow,
                                              float* __restrict__ gsib) {
    __shared__ float Alds[32 * LDS_STRIDE];   // two 16x1024 A tiles (padded rows)
    __shared__ float redG[32];
    __shared__ float redS[32];

    const int it0  = blockIdx.x * 2;          // first i-tile (== batch index)
    const int it1  = it0 + 1;                 // second i-tile
    const int tid  = threadIdx.x;
    const int wave = tid >> 5;
    const int lane = tid & 31;
    const int half = lane >> 4;               // 0: K pair {0,1}, 1: K pair {2,3}
    const int mn   = lane & 15;               // A row m / B col n / C col n

    // Stage both A tiles (32 rows x 1024) into LDS with float4 loads
    {
        const float4* src4 = (const float4*)(nrm + (size_t)it0 * 16 * Dn);
        for (int idx = tid; idx < 32 * Dn / 4; idx += 256) {
            const int r  = idx / (Dn / 4);
            const int c4 = idx % (Dn / 4);
            float4 v = src4[(size_t)r * (Dn / 4) + c4];
            float* drow = &Alds[r * LDS_STRIDE + c4 * 4];
            drow[0] = v.x; drow[1] = v.y; drow[2] = v.z; drow[3] = v.w;
        }
    }
    if (tid < 32) { redG[tid] = 0.0f; redS[tid] = 0.0f; }
    __syncthreads();

    float acc0[8], acc1[8];   // per-lane partial row sums (rows r + 8*half, col slot mn)
    float sac0[8], sac1[8];   // sibling partials (each wave sees each diag tile at most once)
#pragma unroll
    for (int r = 0; r < 8; ++r) { acc0[r] = 0.f; acc1[r] = 0.f; sac0[r] = 0.f; sac1[r] = 0.f; }

    const float* Arow0 = &Alds[mn * LDS_STRIDE];
    const float* Arow1 = &Alds[(16 + mn) * LDS_STRIDE];
    const float* wb0 = w + (size_t)it0 * Vn * Vn;
    const float* wb1 = w + (size_t)it1 * Vn * Vn;

    for (int jt = wave; jt < NTILES; jt += 8) {      // uniform per wave
        const float* Brow = nrm + ((size_t)jt * 16 + mn) * Dn;
        v8f c0 = {0.f, 0.f, 0.f, 0.f, 0.f, 0.f, 0.f, 0.f};
        v8f c1 = {0.f, 0.f, 0.f, 0.f, 0.f, 0.f, 0.f, 0.f};
#pragma unroll 4
        for (int kk = 0; kk < Dn; kk += 4) {
            v2f b  = *(const v2f*)&Brow[kk + 2 * half];   // B = tile_j^T fragment
            v2f a0 = *(const v2f*)&Arow0[kk + 2 * half];  // A: 16x4 fp32 fragments
            v2f a1 = *(const v2f*)&Arow1[kk + 2 * half];
            c0 = __builtin_amdgcn_wmma_f32_16x16x4_f32(
                     false, a0, false, b, (short)0, c0, false, false);
            c1 = __builtin_amdgcn_wmma_f32_16x16x4_f32(
                     false, a1, false, b, (short)0, c1, false, false);
        }
        const float mj = (float)mask[jt * 16 + mn];
        const bool diag0 = (jt == it0);                  // wave-uniform
        const bool diag1 = (jt == it1);
#pragma unroll
        for (int r = 0; r < 8; ++r) {
            const int mrow = r + 8 * half;
            float e0 = expf(c0[r] * INV_T) * mj;         // masked exp(S/T)
            float e1 = expf(c1[r] * INV_T) * mj;
            acc0[r] += e0;
            acc1[r] += e1;
            if (diag0) sac0[r] += wb0[mrow * Vn + mn] * e0;  // w has zero diagonal
            if (diag1) sac1[r] += wb1[mrow * Vn + mn] * e1;
        }
    }

    // Reduce across the 16 lanes of each half (xor stays within half for off<16)
#pragma unroll
    for (int r = 0; r < 8; ++r) {
        float a0 = acc0[r], a1 = acc1[r], s0 = sac0[r], s1 = sac1[r];
#pragma unroll
        for (int off = 1; off < 16; off <<= 1) {
            a0 += __shfl_xor(a0, off, 32);
            a1 += __shfl_xor(a1, off, 32);
            s0 += __shfl_xor(s0, off, 32);
            s1 += __shfl_xor(s1, off, 32);
        }
        if (mn == 0) {                    // lane 0 -> row r, lane 16 -> row r+8
            const int row = r + 8 * half;
            atomicAdd(&redG[row],      a0);   // ds_add_f32 across the 8 waves
            atomicAdd(&redS[row],      s0);
            atomicAdd(&redG[16 + row], a1);
            atomicAdd(&redS[16 + row], s1);
        }
    }
    __syncthreads();

    if (tid < 32) {
        const float self_exp = expf(INV_T);   // exp(1/T), subtracted like reference
        grow[it0 * 16 + tid] = redG[tid] - self_exp;
        gsib[it0 * 16 + tid] = redS[tid];
    }
}

// ---------------------------------------------------------------------------
// Kernel 4: final masked-mean loss
// ---------------------------------------------------------------------------
__global__ void k_final(const float* __restrict__ gpos,
                        const float* __restrict__ grow,
                        const float* __restrict__ gsib,
                        const unsigned char* __restrict__ mask,
                        float* __restrict__ out) {
    __shared__ float redT[256];
    __shared__ float redC[256];
    float tot = 0.0f, cnt = 0.0f;
    for (int i = threadIdx.x; i < BV; i += 256) {
        const float m = (float)mask[i];
        const float p = gpos[i];
        const float denom = gsib[i] + grow[i];
        const float loss = -logf(p / (p + denom + EPS_F));
        tot += loss * m;
        cnt += m;
    }
    redT[threadIdx.x] = tot;
    redC[threadIdx.x] = cnt;
    __syncthreads();
    for (int off = 128; off > 0; off >>= 1) {
        if (threadIdx.x < off) {
            redT[threadIdx.x] += redT[threadIdx.x + off];
            redC[threadIdx.x] += redC[threadIdx.x + off];
        }
        __syncthreads();
    }
    if (threadIdx.x == 0) out[0] = redT[0] / fmaxf(redC[0], 1.0f);
}

// ---------------------------------------------------------------------------
extern "C" void kernel_launch(void* const* d_in, const int* in_sizes, int n_in,
                              void* d_out, int out_size, void* d_ws, size_t ws_size,
                              hipStream_t stream) {
    const float* clean = (const float*)d_in[0];           // [256,1024] f32
    const float* degr  = (const float*)d_in[1];           // [256,16,1024] f32
    const float* w     = (const float*)d_in[2];           // [256,16,16] f32
    const unsigned char* mask = (const unsigned char*)d_in[3]; // [256,16] bool

    float* ws   = (float*)d_ws;
    float* nrm  = ws;                                     // BV*D
    float* cn   = nrm  + (size_t)BV * Dn;                 // B*D
    float* gpos = cn   + (size_t)Bn * Dn;                 // BV
    float* grow = gpos + BV;                              // BV
    float* gsib = grow + BV;                              // BV

    k_normalize<<<BV + Bn, 256, 0, stream>>>(clean, degr, nrm, cn);
    k_pos<<<Bn, 256, 0, stream>>>(nrm, cn, gpos);
    k_gram<<<NTILES / 2, 256, 0, stream>>>(nrm, w, mask, grow, gsib);
    k_final<<<1, 256, 0, stream>>>(gpos, grow, gsib, mask, (float*)d_out);
}